// SaliencyLoss_5042291605815
// MI455X (gfx1250) — compile-verified
//
#include <hip/hip_runtime.h>
#include <stdint.h>

typedef __attribute__((ext_vector_type(2))) float v2f;
typedef __attribute__((ext_vector_type(8))) float v8f;

#define N_POS 8192
#define FEAT  512
#define TILE  128
#define KC    16
#define LDS_STRIDE 18   // 18*i mod 64 distinct for i=0..15 -> conflict-free, keeps 8B align

// Monotonic float->uint mapping; pack (value, ~index) so atomicMax keeps the
// largest value and, on exact ties, the smallest index (matches jnp.argmax).
__device__ __forceinline__ unsigned long long pack_key(float v, unsigned idx) {
    unsigned u = __float_as_uint(v);
    u = (u & 0x80000000u) ? ~u : (u | 0x80000000u);
    return (((unsigned long long)u) << 32) | (unsigned)(~idx);
}

__global__ void init_best_kernel(unsigned long long* best, int n) {
    int i = blockIdx.x * blockDim.x + threadIdx.x;
    if (i < n) best[i] = 0ull;   // 0 == key(-NaN), below any real value
}

__global__ void gather_points_kernel(const float* __restrict__ transform,
                                     const float* __restrict__ src_points,
                                     const float* __restrict__ tgt_points,
                                     const int*   __restrict__ src_idx,
                                     const int*   __restrict__ tgt_idx,
                                     float* __restrict__ src_pp,
                                     float* __restrict__ tgt_pp) {
    int i = blockIdx.x * blockDim.x + threadIdx.x;
    if (i >= N_POS) return;
    int si = src_idx[i];
    float px = src_points[si * 3 + 0];
    float py = src_points[si * 3 + 1];
    float pz = src_points[si * 3 + 2];
#pragma unroll
    for (int j = 0; j < 3; ++j) {
        // points @ R.T + t  with transform row-major 4x4
        src_pp[i * 3 + j] = transform[4 * j + 0] * px + transform[4 * j + 1] * py +
                            transform[4 * j + 2] * pz + transform[4 * j + 3];
    }
    int ti = tgt_idx[i];
    tgt_pp[i * 3 + 0] = tgt_points[ti * 3 + 0];
    tgt_pp[i * 3 + 1] = tgt_points[ti * 3 + 1];
    tgt_pp[i * 3 + 2] = tgt_points[ti * 3 + 2];
}

// 256 threads = 8 wave32; block computes a 128x128 tile of similarity via
// V_WMMA_F32_16X16X4_F32 and merges row/col argmax with packed u64 atomics.
__global__ __launch_bounds__(256)
void sim_argmax_kernel(const float* __restrict__ src_feats,
                       const float* __restrict__ tgt_feats,
                       const int*   __restrict__ src_idx,
                       const int*   __restrict__ tgt_idx,
                       unsigned long long* __restrict__ src_best,
                       unsigned long long* __restrict__ tgt_best) {
    __shared__ float ldsT[TILE * LDS_STRIDE];

    const int tid  = threadIdx.x;
    const int wave = tid >> 5;
    const int lane = tid & 31;
    const int half = lane >> 4;     // K-half for A/B frags, M-half for C
    const int nIn  = lane & 15;

    const int tileM = blockIdx.y * TILE;
    const int tileN = blockIdx.x * TILE;

    // A fragment row: gathered src row for this lane (fixed for whole K loop)
    const long aRow = (long)src_idx[tileM + wave * 16 + nIn] * FEAT;

    // Staging assignment: thread stages rows sr0 and sr0+64, K-chunk sc0..sc0+3
    const int  sr0   = tid >> 2;
    const int  sc0   = (tid & 3) * 4;
    const long tRow0 = (long)tgt_idx[tileN + sr0]      * FEAT;
    const long tRow1 = (long)tgt_idx[tileN + sr0 + 64] * FEAT;

    v8f c[8] = {};   // 16 (M) x 128 (N) accumulators per wave

    for (int k0 = 0; k0 < FEAT; k0 += KC) {
        // Stage tgt tile [128 x KC] into padded LDS
        {
            float4 t0 = *(const float4*)(tgt_feats + tRow0 + k0 + sc0);
            float4 t1 = *(const float4*)(tgt_feats + tRow1 + k0 + sc0);
            float* p0 = &ldsT[sr0 * LDS_STRIDE + sc0];
            p0[0] = t0.x; p0[1] = t0.y; p0[2] = t0.z; p0[3] = t0.w;
            float* p1 = &ldsT[(sr0 + 64) * LDS_STRIDE + sc0];
            p1[0] = t1.x; p1[1] = t1.y; p1[2] = t1.z; p1[3] = t1.w;
        }
        __syncthreads();
#pragma unroll
        for (int kk = 0; kk < KC; kk += 4) {
            const int koff = kk + half * 2;                  // lanes 0-15: K=0,1  lanes 16-31: K=2,3
            v2f a = *(const v2f*)(src_feats + aRow + k0 + koff);
#pragma unroll
            for (int nt = 0; nt < 8; ++nt) {
                v2f b = *(const v2f*)(&ldsT[(nt * 16 + nIn) * LDS_STRIDE + koff]);
                c[nt] = __builtin_amdgcn_wmma_f32_16x16x4_f32(
                    /*neg_a=*/false, a, /*neg_b=*/false, b,
                    /*c_mod=*/(short)0, c[nt], /*reuse_a=*/false, /*reuse_b=*/false);
            }
        }
        __syncthreads();
    }

    // ---- Row argmax over this block's 128 columns ----
    // C layout: lane holds N = nIn; VGPR j holds M = j + half*8
#pragma unroll
    for (int j = 0; j < 8; ++j) {
        float    bv = c[0][j];
        unsigned bc = (unsigned)(tileN + nIn);
#pragma unroll
        for (int nt = 1; nt < 8; ++nt) {
            float v = c[nt][j];
            unsigned cc = (unsigned)(tileN + nt * 16 + nIn);
            if (v > bv) { bv = v; bc = cc; }       // strict > keeps first (smaller) col on ties
        }
        unsigned long long key = pack_key(bv, bc);
#pragma unroll
        for (int m = 1; m <= 8; m <<= 1) {         // reduce the 16 lanes of this half
            unsigned long long o = __shfl_xor(key, m, 32);
            if (o > key) key = o;
        }
        if (nIn == 0)
            atomicMax(&src_best[tileM + wave * 16 + half * 8 + j], key);
    }

    // ---- Col argmax over this wave's 16 rows ----
#pragma unroll
    for (int nt = 0; nt < 8; ++nt) {
        float    bv = c[nt][0];
        unsigned br = (unsigned)(tileM + wave * 16 + half * 8);
#pragma unroll
        for (int j = 1; j < 8; ++j) {
            float v = c[nt][j];
            if (v > bv) { bv = v; br = (unsigned)(tileM + wave * 16 + half * 8 + j); }
        }
        unsigned long long key = pack_key(bv, br);
        unsigned long long o = __shfl_xor(key, 16, 32);   // merge the two M halves
        if (o > key) key = o;
        if (half == 0)
            atomicMax(&tgt_best[tileN + nt * 16 + nIn], key);
    }
}

__device__ __forceinline__ void item_eval(int i,
        const unsigned long long* sb, const unsigned long long* tb,
        const float* spp, const float* tpp,
        const float* ss, const float* ts,
        const int* si, const int* ti,
        float& label, float& score) {
    if (i < N_POS) {
        unsigned nn = ~(unsigned)sb[i];
        float dx = spp[3 * i + 0] - tpp[3 * nn + 0];
        float dy = spp[3 * i + 1] - tpp[3 * nn + 1];
        float dz = spp[3 * i + 2] - tpp[3 * nn + 2];
        label = (sqrtf(dx * dx + dy * dy + dz * dz) < 0.1f) ? 1.0f : 0.0f;
        score = ss[si[i]];
    } else {
        int j = i - N_POS;
        unsigned nn = ~(unsigned)tb[j];
        float dx = tpp[3 * j + 0] - spp[3 * nn + 0];
        float dy = tpp[3 * j + 1] - spp[3 * nn + 1];
        float dz = tpp[3 * j + 2] - spp[3 * nn + 2];
        label = (sqrtf(dx * dx + dy * dy + dz * dz) < 0.1f) ? 1.0f : 0.0f;
        score = ts[ti[j]];
    }
}

__device__ __forceinline__ float block_sum(float v, float* red) {
    int tid = threadIdx.x;
    red[tid] = v;
    __syncthreads();
    for (int s = 128; s > 0; s >>= 1) {
        if (tid < s) red[tid] += red[tid + s];
        __syncthreads();
    }
    float r = red[0];
    __syncthreads();
    return r;
}

__global__ void finalize_kernel(const unsigned long long* __restrict__ src_best,
                                const unsigned long long* __restrict__ tgt_best,
                                const float* __restrict__ src_pp,
                                const float* __restrict__ tgt_pp,
                                const float* __restrict__ src_scores,
                                const float* __restrict__ tgt_scores,
                                const int*   __restrict__ src_idx,
                                const int*   __restrict__ tgt_idx,
                                float* __restrict__ out) {
    __shared__ float red[256];
    const int tid = threadIdx.x;

    float sumLab = 0.f, sumRes = 0.f, corr = 0.f;
    for (int i = tid; i < 2 * N_POS; i += 256) {
        float label, score;
        item_eval(i, src_best, tgt_best, src_pp, tgt_pp,
                  src_scores, tgt_scores, src_idx, tgt_idx, label, score);
        float res = (score > 0.5f) ? 1.0f : 0.0f;
        sumLab += label;
        sumRes += res;
        corr   += res * label;
    }
    sumLab = block_sum(sumLab, red);
    sumRes = block_sum(sumRes, red);
    corr   = block_sum(corr,   red);

    const float nw = sumLab / (float)(2 * N_POS);   // negative_weight = mean(targets)
    const float pw = 1.0f - nw;

    float wsum = 0.f;
    for (int i = tid; i < 2 * N_POS; i += 256) {
        float label, score;
        item_eval(i, src_best, tgt_best, src_pp, tgt_pp,
                  src_scores, tgt_scores, src_idx, tgt_idx, label, score);
        float w   = label * nw + (1.0f - label) * pw;
        float bce = -(label * logf(score) + (1.0f - label) * logf(1.0f - score));
        wsum += w * bce;
    }
    wsum = block_sum(wsum, red);

    if (tid == 0) {
        out[0] = wsum / (float)(2 * N_POS);          // loss (LOSS_WEIGHT = 1)
        out[1] = corr / (sumRes + 1e-12f);           // precision
        out[2] = corr / (sumLab + 1e-12f);           // recall
    }
}

extern "C" void kernel_launch(void* const* d_in, const int* in_sizes, int n_in,
                              void* d_out, int out_size, void* d_ws, size_t ws_size,
                              hipStream_t stream) {
    const float* transform  = (const float*)d_in[0];
    const float* src_points = (const float*)d_in[1];
    const float* tgt_points = (const float*)d_in[2];
    const float* src_scores = (const float*)d_in[3];
    const float* tgt_scores = (const float*)d_in[4];
    const float* src_feats  = (const float*)d_in[5];
    const float* tgt_feats  = (const float*)d_in[6];
    const int*   src_idx    = (const int*)d_in[7];
    const int*   tgt_idx    = (const int*)d_in[8];
    float*       out        = (float*)d_out;

    char* ws = (char*)d_ws;
    unsigned long long* src_best = (unsigned long long*)ws;                // 64 KB
    unsigned long long* tgt_best = (unsigned long long*)(ws + 65536);      // 64 KB
    float* src_pp = (float*)(ws + 131072);                                 // 96 KB
    float* tgt_pp = (float*)(ws + 131072 + 98304);                         // 96 KB

    init_best_kernel<<<64, 256, 0, stream>>>(src_best, 2 * N_POS);
    gather_points_kernel<<<32, 256, 0, stream>>>(transform, src_points, tgt_points,
                                                 src_idx, tgt_idx, src_pp, tgt_pp);
    dim3 grid(TILE == 128 ? N_POS / TILE : 64, N_POS / TILE);
    sim_argmax_kernel<<<grid, 256, 0, stream>>>(src_feats, tgt_feats,
                                                src_idx, tgt_idx, src_best, tgt_best);
    finalize_kernel<<<1, 256, 0, stream>>>(src_best, tgt_best, src_pp, tgt_pp,
                                           src_scores, tgt_scores, src_idx, tgt_idx, out);
}